// Snake_39402029973565
// MI455X (gfx1250) — compile-verified
//
#include <hip/hip_runtime.h>
#include <stdint.h>
#include <stddef.h>

static constexpr int Hc    = 8192;
static constexpr int Wc    = 2048;
static constexpr int Pc    = 6;      // NUM_PEAKS
static constexpr int Kc    = 12;     // 2*P
static constexpr int HALFc = 22;     // MIN_DIST // 2
static constexpr int CAPc  = 16;     // per-column candidate capacity
static constexpr int CH    = 1024;   // floats staged per block iteration (256 lanes * 4)

__device__ __forceinline__ float f32_nan()  { return __int_as_float(0x7fc00000); }
__device__ __forceinline__ float f32_inf()  { return __int_as_float(0x7f800000); }
__device__ __forceinline__ float f32_ninf() { return __int_as_float(0xff800000); }

typedef int v4i_t __attribute__((vector_size(16)));

// ---------------- CDNA5 async global->LDS staging ----------------
__device__ __forceinline__ void async_ld16(const float* g, float* l) {
#if __has_builtin(__builtin_amdgcn_global_load_async_to_lds_b128)
  __builtin_amdgcn_global_load_async_to_lds_b128(
      (__attribute__((address_space(1))) v4i_t*)(uintptr_t)g,
      (__attribute__((address_space(3))) v4i_t*)(uint32_t)(uintptr_t)l,
      0, 0);
#else
  uint32_t lds = (uint32_t)(uintptr_t)l;
  asm volatile("global_load_async_to_lds_b128 %0, %1, off"
               :: "v"(lds), "v"(g) : "memory");
#endif
}

__device__ __forceinline__ void wait_async_le0() {
#if __has_builtin(__builtin_amdgcn_s_wait_asynccnt)
  __builtin_amdgcn_s_wait_asynccnt(0);
#else
  asm volatile("s_wait_asynccnt 0" ::: "memory");
#endif
  asm volatile("" ::: "memory");
}

__device__ __forceinline__ void wait_async_le1() {
#if __has_builtin(__builtin_amdgcn_s_wait_asynccnt)
  __builtin_amdgcn_s_wait_asynccnt(1);
#else
  asm volatile("s_wait_asynccnt 1" ::: "memory");
#endif
  asm volatile("" ::: "memory");
}

// ---------------- kernel 0: zero scratch header ----------------
__global__ void k_init(int* p, int n) {
  int i = blockIdx.x * blockDim.x + threadIdx.x;
  if (i < n) p[i] = 0;
}

// ---------------- kernel 1: global max ----------------
__global__ void __launch_bounds__(256) k_gmax(const float4* __restrict__ p4, int n4, int* gmax) {
  float m = 0.0f;
  for (int i = blockIdx.x * blockDim.x + threadIdx.x; i < n4; i += gridDim.x * blockDim.x) {
    float4 v = p4[i];
    m = fmaxf(m, fmaxf(fmaxf(v.x, v.y), fmaxf(v.z, v.w)));
  }
  __shared__ float red[256];
  red[threadIdx.x] = m;
  __syncthreads();
  for (int s = 128; s > 0; s >>= 1) {
    if (threadIdx.x < s) red[threadIdx.x] = fmaxf(red[threadIdx.x], red[threadIdx.x + s]);
    __syncthreads();
  }
  if (threadIdx.x == 0) atomicMax(gmax, __float_as_int(red[0]));
}

// ---------------- kernel 2: threshold-first local-max candidate extraction ----------------
__global__ void __launch_bounds__(256) k_peaks(const float* __restrict__ preds,
                                               const int* __restrict__ gmax,
                                               int* __restrict__ cnt,
                                               float* __restrict__ cv,
                                               int* __restrict__ cr) {
  __shared__ float stage[2][CH];
  const float thr = 0.5f * __int_as_float(*gmax);
  const int tid = threadIdx.x;
  const int nChunks = (Hc * Wc) / CH;
  int ch = blockIdx.x;
  const int stride = gridDim.x;
  int buf = 0;
  if (ch < nChunks)
    async_ld16(preds + (size_t)ch * CH + tid * 4, &stage[0][tid * 4]);
  for (; ch < nChunks; ch += stride) {
    int nxt = ch + stride;
    if (nxt < nChunks) {
      async_ld16(preds + (size_t)nxt * CH + tid * 4, &stage[buf ^ 1][tid * 4]);
      wait_async_le1();
    } else {
      wait_async_le0();
    }
#pragma unroll
    for (int u = 0; u < 4; ++u) {
      float v = stage[buf][tid * 4 + u];
      int idx = ch * CH + tid * 4 + u;
      if (v >= thr) {
        int r = idx >> 11;
        int c = idx & (Wc - 1);
        int lo = r - HALFc; if (lo < 0) lo = 0;
        int hi = r + HALFc; if (hi > Hc - 1) hi = Hc - 1;
        float wm = f32_ninf();
        for (int rr = lo; rr <= hi; ++rr) wm = fmaxf(wm, preds[(size_t)rr * Wc + c]);
        if (v == wm) {
          int pos = atomicAdd(&cnt[c], 1);
          if (pos < CAPc) { cv[c * CAPc + pos] = v; cr[c * CAPc + pos] = r; }
        }
      }
    }
    buf ^= 1;
  }
}

// ---------------- kernel 3: per-column stable top-12 (value desc, row asc) ----------------
__global__ void __launch_bounds__(256) k_topk(const int* __restrict__ cnt,
                                              const float* __restrict__ cv,
                                              const int* __restrict__ cr,
                                              float* __restrict__ okbuf,
                                              int* __restrict__ kcap) {
  int c = blockIdx.x * blockDim.x + threadIdx.x;
  if (c >= Wc) return;
  int n = cnt[c];
  int nn = n < CAPc ? n : CAPc;
  float bv[Kc];
  int   br[Kc];
#pragma unroll
  for (int j = 0; j < Kc; ++j) { bv[j] = f32_ninf(); br[j] = 0x7fffffff; }
  for (int t = 0; t < nn; ++t) {
    float v = cv[c * CAPc + t];
    int   r = cr[c * CAPc + t];
#pragma unroll
    for (int j = 0; j < Kc; ++j) {
      bool ins = (v > bv[j]) || (v == bv[j] && r < br[j]);
      float tv = bv[j]; int tr = br[j];
      bv[j] = ins ? v : tv;  br[j] = ins ? r : tr;
      v     = ins ? tv : v;  r     = ins ? tr : r;
    }
  }
#pragma unroll
  for (int j = 0; j < Kc; ++j) okbuf[j * Wc + c] = (float)br[j];
  kcap[c] = n < Kc ? n : Kc;
}

// ---------------- shared column-update math (replicates JAX body exactly) ----------------
template <int N>
__device__ __forceinline__ void sortAsc(float* a) {
#pragma unroll
  for (int x = 0; x < N; ++x)
#pragma unroll
    for (int y = 0; y < N - 1; ++y)
      if (a[y] > a[y + 1]) { float t = a[y]; a[y] = a[y + 1]; a[y + 1] = t; }
}

// other: 6 raw entries of ptop[:,idx]; cur: 12 raw (valid-masked) entries; srt: 6 raw of ptop[:,i]
__device__ __forceinline__ void compute_newcol(int k, bool found,
                                               float* other, float* cur, float* srt,
                                               float* ncol) {
  const float INF = f32_inf();
  sortAsc<Pc>(other);
  sortAsc<Kc>(cur);
  sortAsc<Pc>(srt);

  // greedy (k > P): P sequential first-occurrence argmins, j<=prev masked (prev may decrease)
  float greedy[Pc];
  int prev = -1;
#pragma unroll
  for (int r = 0; r < Pc; ++r) {
    float best = INF; int bj = 0;
#pragma unroll
    for (int j = 0; j < Kc; ++j) {
      float d = (j <= prev) ? INF : fabsf(other[r] - cur[j]);
      if (d < best) { best = d; bj = j; }
    }
    float pick = INF;
#pragma unroll
    for (int j = 0; j < Kc; ++j) pick = (j == bj) ? cur[j] : pick;
    greedy[r] = pick;
    prev = bj;
  }

  // scatter (k <= P): last-write-wins in ascending j
  float sc[Pc];
#pragma unroll
  for (int p = 0; p < Pc; ++p) sc[p] = f32_nan();
#pragma unroll
  for (int j = 0; j < Kc; ++j) {
    float best = INF; int rb = 0;
#pragma unroll
    for (int r = 0; r < Pc; ++r) {
      float d = fabsf(other[r] - cur[j]);
      if (d < best) { best = d; rb = r; }
    }
    int tgt = (j < k) ? rb : Pc;
#pragma unroll
    for (int p = 0; p < Pc; ++p) if (tgt == p) sc[p] = cur[j];
  }

#pragma unroll
  for (int p = 0; p < Pc; ++p) {
    float al = (k > Pc) ? greedy[p] : sc[p];
    ncol[p] = (k == Pc) ? srt[p]
             : ((k >= Kc) ? f32_nan() : (found ? al : f32_nan()));
  }
}

// ---------------- kernel 4: column scan (parallel fast path / exact sequential fallback) ----------------
__global__ void __launch_bounds__(256) k_scan(const float* __restrict__ okbuf,
                                              const int* __restrict__ kcap,
                                              int* __restrict__ wnz,
                                              float* __restrict__ wptop,
                                              float* __restrict__ out) {
  __shared__ int sprev[Wc];   // nearest c'<=c with nz0==P, else -1
  __shared__ int snext[Wc];   // nearest c'>=c with nz0==P, else Wc
  __shared__ int r0[256], r1[256], r2[256], r3[256];
  const int tid = threadIdx.x;
  constexpr int COLS = Wc / 256;  // 8 columns per thread

  // ---- flags + epilogue stats from nz0 (kcap) ----
  int pcnt = 0, pfirst = Wc, plast = -1, bits = 0;
#pragma unroll
  for (int u = 0; u < COLS; ++u) {
    int c = tid + u * 256;
    int k = kcap[c];
    if (k == Pc) { pcnt++; if (c < pfirst) pfirst = c; if (c > plast) plast = c; bits |= 1; }
    if (k > Pc && k < Kc) bits |= 2;   // anyMid: sequential fallback needed
    sprev[c] = (k == Pc) ? c : -1;
    snext[c] = (k == Pc) ? c : Wc;
  }
  r0[tid] = pcnt; r1[tid] = pfirst; r2[tid] = plast; r3[tid] = bits;
  __syncthreads();
  for (int s = 128; s > 0; s >>= 1) {
    if (tid < s) {
      r0[tid] += r0[tid + s];
      r1[tid] = r1[tid] < r1[tid + s] ? r1[tid] : r1[tid + s];
      r2[tid] = r2[tid] > r2[tid + s] ? r2[tid] : r2[tid + s];
      r3[tid] |= r3[tid + s];
    }
    __syncthreads();
  }
  const int cntm = r0[0];
  int first = r1[0], last = r2[0];
  const bool anyP  = (r3[0] & 1) != 0;
  const bool anyMid = (r3[0] & 2) != 0;
  if (cntm < 2) { first = -1; last = 0; }
  const int left = (first < 0) ? (Wc - 1) : first;

  // ---- nearest-P arrays: Hillis-Steele max/min propagation ----
  for (int off = 1; off < Wc; off <<= 1) {
    int tp[COLS], tn[COLS];
#pragma unroll
    for (int u = 0; u < COLS; ++u) {
      int c = tid + u * 256;
      int pv = sprev[c]; if (c >= off)      { int o = sprev[c - off]; if (o > pv) pv = o; }
      int nv = snext[c]; if (c + off < Wc)  { int o = snext[c + off]; if (o < nv) nv = o; }
      tp[u] = pv; tn[u] = nv;
    }
    __syncthreads();
#pragma unroll
    for (int u = 0; u < COLS; ++u) { int c = tid + u * 256; sprev[c] = tp[u]; snext[c] = tn[u]; }
    __syncthreads();
  }

  if (!anyMid) {
    // ===== parallel exact path: nz static; every referenced ptop column is original up to a sort =====
    for (int u = 0; u < COLS; ++u) {
      int c = tid + u * 256;
      int k = kcap[c];
      int cL = sprev[c], cR = snext[c];
      int idx;
      if (cL == c) idx = c;  // key 0 at self
      else {
        int kl = (cL >= 0) ? (2 * (c - cL) - 1) : 0x7fffffff;
        int kr = (cR < Wc) ? (2 * (cR - c))     : 0x7fffffff;
        idx = (kl == 0x7fffffff && kr == 0x7fffffff) ? 0 : ((kl < kr) ? cL : cR);
      }
      float other[Pc], srt[Pc], cur[Kc], ncol[Pc];
#pragma unroll
      for (int p = 0; p < Pc; ++p) { other[p] = okbuf[p * Wc + idx]; srt[p] = okbuf[p * Wc + c]; }
#pragma unroll
      for (int j = 0; j < Kc; ++j) cur[j] = (j < k) ? okbuf[j * Wc + c] : f32_inf();
      compute_newcol(k, anyP, other, cur, srt, ncol);
      bool m = (c < left) || (c > last);
#pragma unroll
      for (int p = 0; p < Pc; ++p) out[p * Wc + c] = m ? f32_nan() : ncol[p];
    }
  } else {
    // ===== exact sequential fallback: single thread, O(1) nearest query, state in workspace =====
    if (tid == 0) {
      for (int c = 0; c < Wc; ++c) {
        wnz[c] = kcap[c];
#pragma unroll
        for (int p = 0; p < Pc; ++p) wptop[p * Wc + c] = okbuf[p * Wc + c];
      }
      int lastLeft = -1;
      for (int i = 0; i < Wc; ++i) {
        int k = wnz[i];                       // == kcap[i]: own slot untouched before iteration i
        int nR = snext[i];                    // right side keeps original nz
        bool found = (lastLeft >= 0) || (nR < Wc);
        int idx;
        if (k == Pc) idx = i;
        else {
          int kl = (lastLeft >= 0) ? (2 * (i - lastLeft) - 1) : 0x7fffffff;
          int kr = (nR < Wc)       ? (2 * (nR - i))           : 0x7fffffff;
          idx = (kl == 0x7fffffff && kr == 0x7fffffff) ? 0 : ((kl < kr) ? lastLeft : nR);
        }
        float other[Pc], srt[Pc], cur[Kc], ncol[Pc];
#pragma unroll
        for (int p = 0; p < Pc; ++p) { other[p] = wptop[p * Wc + idx]; srt[p] = wptop[p * Wc + i]; }
#pragma unroll
        for (int j = 0; j < Kc; ++j) cur[j] = (j < k) ? okbuf[j * Wc + i] : f32_inf();
        compute_newcol(k, found, other, cur, srt, ncol);
#pragma unroll
        for (int p = 0; p < Pc; ++p) wptop[p * Wc + i] = ncol[p];
        bool fixed = (k > Pc) && (k < Kc) && found;
        int nzNew = fixed ? Pc : k;
        wnz[i] = nzNew;
        if (nzNew == Pc) lastLeft = i;
      }
      for (int c = 0; c < Wc; ++c) {
        bool m = (c < left) || (c > last);
#pragma unroll
        for (int p = 0; p < Pc; ++p) out[p * Wc + c] = m ? f32_nan() : wptop[p * Wc + c];
      }
    }
  }
}

// ---------------- launcher ----------------
extern "C" void kernel_launch(void* const* d_in, const int* in_sizes, int n_in,
                              void* d_out, int out_size, void* d_ws, size_t ws_size,
                              hipStream_t stream) {
  (void)in_sizes; (void)n_in; (void)out_size; (void)ws_size;
  const float* preds = (const float*)d_in[0];
  float* out = (float*)d_out;
  char* ws = (char*)d_ws;

  size_t o = 0;
  int*   gmax  = (int*)(ws + o);   o += 256;
  int*   cnt   = (int*)(ws + o);   o += Wc * 4;
  float* cv    = (float*)(ws + o); o += (size_t)Wc * CAPc * 4;
  int*   cr    = (int*)(ws + o);   o += (size_t)Wc * CAPc * 4;
  float* okbuf = (float*)(ws + o); o += (size_t)Kc * Wc * 4;
  int*   kcap  = (int*)(ws + o);   o += Wc * 4;
  int*   wnz   = (int*)(ws + o);   o += Wc * 4;
  float* wptop = (float*)(ws + o); o += (size_t)Pc * Wc * 4;

  const int initInts = (256 + Wc * 4) / 4;   // gmax pad + cnt
  k_init<<<(initInts + 255) / 256, 256, 0, stream>>>((int*)ws, initInts);
  k_gmax<<<2048, 256, 0, stream>>>((const float4*)preds, (Hc * Wc) / 4, gmax);
  k_peaks<<<2048, 256, 0, stream>>>(preds, gmax, cnt, cv, cr);
  k_topk<<<Wc / 256, 256, 0, stream>>>(cnt, cv, cr, okbuf, kcap);
  k_scan<<<1, 256, 0, stream>>>(okbuf, kcap, wnz, wptop, out);
}